// FlowEmbedding_12008728560017
// MI455X (gfx1250) — compile-verified
//
#include <hip/hip_runtime.h>

// ---------------------------------------------------------------------------
// FlowEmbedding for MI455X (gfx1250, wave32).
// Pipeline (all on `stream`):
//   1. knn_kernel                : brute-force 32-NN + radius filter -> idx
//   2. mlp_gemm_kernel<0>        : gather X0 on the fly, Y0 = W0*X0 (fp32 WMMA),
//                                  per-block partial sum/sumsq
//   3. reduce_stats_kernel       : deterministic reduction -> scale0/shift0
//   4. mlp_gemm_kernel<1>        : bn0+relu fused into staging, Y1 = W1*A0
//   5. reduce_stats_kernel       : -> scale1/shift1
//   6. mlp_gemm_kernel<2>        : bn1+relu, Y2 = W2*A1
//   7. reduce_stats_kernel       : -> scale2/shift2
//   8. maxpool_kernel            : bn2+relu+max over S -> feat output
//   9. copy_pos_kernel           : pos1 passthrough (first tuple element)
// ---------------------------------------------------------------------------

typedef __attribute__((ext_vector_type(2))) float v2f;
typedef __attribute__((ext_vector_type(8))) float v8f;

static constexpr int Bsz  = 4;
static constexpr int Npts = 4096;
static constexpr int Mpts = 4096;
static constexpr int Cch  = 64;
static constexpr int Ssz  = 32;
static constexpr int Ttot = Bsz * Npts * Ssz;     // 524288 columns
static constexpr int NBLK = Ttot / 32;            // 16384 gemm blocks (== B*N)
static constexpr float RAD2 = 25.0f;              // RADIUS^2
static constexpr float BNEPS = 1e-5f;

// ---------------------------------------------------------------------------
// 1. KNN: one thread per query point, sorted-ascending top-32 heap in LDS.
// ---------------------------------------------------------------------------
#define KNN_BLK 128

__global__ __launch_bounds__(KNN_BLK)
void knn_kernel(const float* __restrict__ pos1, const float* __restrict__ pos2,
                int* __restrict__ idxOut)
{
    // per-thread heaps, strided layout so adjacent lanes hit distinct banks
    __shared__ float dh[32 * KNN_BLK];
    __shared__ int   ih[32 * KNN_BLK];

    const int tid = threadIdx.x;
    const int b   = blockIdx.x / (Npts / KNN_BLK);
    const int n   = (blockIdx.x % (Npts / KNN_BLK)) * KNN_BLK + tid;

    const float* p2b = pos2 + b * 3 * Mpts;      // pos2[b] stays hot in L2/WGP$
    const float p1x = pos1[b * 3 * Npts + 0 * Npts + n];
    const float p1y = pos1[b * 3 * Npts + 1 * Npts + n];
    const float p1z = pos1[b * 3 * Npts + 2 * Npts + n];

    int   cnt   = 0;
    float worst = 3.4e38f;

    for (int m = 0; m < Mpts; ++m) {
        float dx = p2b[m] - p1x;
        float dy = p2b[Mpts + m] - p1y;
        float dz = p2b[2 * Mpts + m] - p1z;
        float d  = dx * dx + dy * dy + dz * dz;
        if (cnt < 32) {
            int p = cnt;
            while (p > 0 && dh[(p - 1) * KNN_BLK + tid] > d) {
                dh[p * KNN_BLK + tid] = dh[(p - 1) * KNN_BLK + tid];
                ih[p * KNN_BLK + tid] = ih[(p - 1) * KNN_BLK + tid];
                --p;
            }
            dh[p * KNN_BLK + tid] = d;
            ih[p * KNN_BLK + tid] = m;
            ++cnt;
            if (cnt == 32) worst = dh[31 * KNN_BLK + tid];
        } else if (d < worst) {
            int p = 31;
            while (p > 0 && dh[(p - 1) * KNN_BLK + tid] > d) {
                dh[p * KNN_BLK + tid] = dh[(p - 1) * KNN_BLK + tid];
                ih[p * KNN_BLK + tid] = ih[(p - 1) * KNN_BLK + tid];
                --p;
            }
            dh[p * KNN_BLK + tid] = d;
            ih[p * KNN_BLK + tid] = m;
            worst = dh[31 * KNN_BLK + tid];
        }
    }

    const int j0   = ih[tid];                 // nearest neighbor (sorted asc)
    const int base = (b * Npts + n) * 32;
    for (int s = 0; s < 32; ++s) {
        float d = dh[s * KNN_BLK + tid];
        int   j = ih[s * KNN_BLK + tid];
        idxOut[base + s] = (d > RAD2) ? j0 : j;
    }
}

// ---------------------------------------------------------------------------
// 2/4/6. Fused stage + GEMM (fp32 WMMA 16x16x4) + partial BN stats.
// Block = 256 threads = 8 waves: waves laid out 4 (o-tiles) x 2 (t-subtiles);
// block output tile = 64 channels x 32 columns. All 32 columns share (b,n).
//
// WMMA f32 16x16x4 fragment layout (ISA 7.12.2):
//   A (16x4):  vgpr pair = (K=2*half, K=2*half+1) for row M = lane&15
//   B (4x16):  vgpr pair = (K=2*half, K=2*half+1) for col N = lane&15
//   C/D v8f:   elem r -> (M = r + 8*half, N = lane&15)
// ---------------------------------------------------------------------------
template <int LAYER>
__global__ __launch_bounds__(256)
void mlp_gemm_kernel(const float* __restrict__ Wsrc,
                     const float* __restrict__ pos1, const float* __restrict__ pos2,
                     const float* __restrict__ f1,   const float* __restrict__ f2,
                     const int*   __restrict__ idx,
                     const float* __restrict__ Yprev,
                     const float* __restrict__ scaleIn, const float* __restrict__ shiftIn,
                     float* __restrict__ Yout,
                     float* __restrict__ gSum, float* __restrict__ gSq)   // [64][NBLK]
{
    constexpr int KSRC = (LAYER == 0) ? 131 : 64;   // real input channels
    constexpr int KD   = (LAYER == 0) ? 132 : 64;   // zero-padded to mult of 4
    constexpr int XST  = 136;                       // Xs row stride (floats)

    __shared__ float Ws[64 * 132];          // zero-padded weights
    __shared__ float Xs[32 * XST];          // staged X tile: [s][c]
    __shared__ float sPart[2][2][64];       // [sum|sumsq][tsub][channel]
    __shared__ int   sIdx[32];

    const int tid = threadIdx.x;
    const int blk = blockIdx.x;             // == b*N + n
    const int t0  = blk * 32;
    const int b   = blk >> 12;
    const int n   = blk & (Npts - 1);

    // -- stage weights (zero-pad column KSRC..KD-1) --
    for (int i = tid; i < 64 * KD; i += 256) {
        int o = i / KD;
        int c = i - o * KD;
        Ws[i] = (c < KSRC) ? Wsrc[o * KSRC + c] : 0.0f;
    }
    if (LAYER == 0) {
        if (tid < 32) sIdx[tid] = idx[t0 + tid];
    }
    __syncthreads();

    // -- stage X tile --
    if (LAYER == 0) {
        const float* p1b = pos1 + b * 3 * Npts;
        const float* p2b = pos2 + b * 3 * Npts;
        const float* f1b = f1 + b * Cch * Npts;
        const float* f2b = f2 + b * Cch * Npts;
        for (int i = tid; i < 32 * 132; i += 256) {
            int s = i & 31;
            int c = i >> 5;
            float v;
            if (c < 3)        v = p2b[c * Npts + sIdx[s]] - p1b[c * Npts + n];
            else if (c < 67)  v = f2b[(c - 3) * Npts + sIdx[s]];
            else if (c < 131) v = f1b[(c - 67) * Npts + n];
            else              v = 0.0f;
            Xs[s * XST + c] = v;
        }
    } else {
        for (int i = tid; i < 32 * 64; i += 256) {
            int s = i & 31;
            int c = i >> 5;
            float y = Yprev[c * Ttot + t0 + s];
            float v = y * scaleIn[c] + shiftIn[c];      // BN affine (prev layer)
            Xs[s * XST + c] = fmaxf(v, 0.0f);           // ReLU
        }
    }
    __syncthreads();

    const int lane  = tid & 31;
    const int wid   = tid >> 5;
    const int otile = wid & 3;
    const int tsub  = wid >> 2;
    const int lh    = lane >> 4;     // half-wave id
    const int lm    = lane & 15;
    const int tcol  = tsub * 16;

    const float* wrow = &Ws[(otile * 16 + lm) * KD + lh * 2];
    const float* xrow = &Xs[(tcol + lm) * XST + lh * 2];

    v8f acc = {};
    for (int k0 = 0; k0 < KD; k0 += 4) {
        v2f a = *(const v2f*)(wrow + k0);       // aligned ds_load_b64
        v2f x = *(const v2f*)(xrow + k0);       // aligned ds_load_b64
        acc = __builtin_amdgcn_wmma_f32_16x16x4_f32(
                  false, a, false, x, (short)0, acc, false, false);
    }

    // -- epilogue: store pre-BN Y, deterministic per-block channel stats --
    const int tg = t0 + tcol + lm;
#pragma unroll
    for (int r = 0; r < 8; ++r) {
        int   o = otile * 16 + 8 * lh + r;
        float v = acc[r];
        Yout[o * Ttot + tg] = v;
        float s1 = v, s2 = v * v;
#pragma unroll
        for (int msk = 1; msk < 16; msk <<= 1) {      // reduce over 16 columns
            s1 += __shfl_xor(s1, msk, 32);
            s2 += __shfl_xor(s2, msk, 32);
        }
        if (lm == 0) {                                 // unique (tsub,o) writer
            sPart[0][tsub][o] = s1;
            sPart[1][tsub][o] = s2;
        }
    }
    __syncthreads();
    if (tid < 64) {
        gSum[tid * NBLK + blk] = sPart[0][0][tid] + sPart[0][1][tid];
        gSq [tid * NBLK + blk] = sPart[1][0][tid] + sPart[1][1][tid];
    }
}

// ---------------------------------------------------------------------------
// 3/5/7. Deterministic stats reduction -> BN (scale, shift) per channel.
// One block per channel; fixed-order strided sum + fixed binary tree.
// ---------------------------------------------------------------------------
__global__ __launch_bounds__(256)
void reduce_stats_kernel(const float* __restrict__ gSum, const float* __restrict__ gSq,
                         const float* __restrict__ gamma, const float* __restrict__ beta,
                         float* __restrict__ scale, float* __restrict__ shift)
{
    __shared__ float r1[256], r2[256];
    const int o   = blockIdx.x;
    const int tid = threadIdx.x;
    float a = 0.0f, q = 0.0f;
    for (int i = tid; i < NBLK; i += 256) {
        a += gSum[o * NBLK + i];
        q += gSq [o * NBLK + i];
    }
    r1[tid] = a; r2[tid] = q;
    __syncthreads();
    for (int st = 128; st > 0; st >>= 1) {
        if (tid < st) { r1[tid] += r1[tid + st]; r2[tid] += r2[tid + st]; }
        __syncthreads();
    }
    if (tid == 0) {
        const float inv  = 1.0f / (float)Ttot;
        float mean = r1[0] * inv;
        float var  = r2[0] * inv - mean * mean;       // population var (ddof=0)
        float rs   = rsqrtf(var + BNEPS);
        float sc   = gamma[o] * rs;
        scale[o] = sc;
        shift[o] = beta[o] - mean * sc;
    }
}

// ---------------------------------------------------------------------------
// 8. BN2 + ReLU + max over S. One thread per (b,o,n); 8x b128 loads.
// ---------------------------------------------------------------------------
__global__ __launch_bounds__(256)
void maxpool_kernel(const float* __restrict__ Y,
                    const float* __restrict__ scale, const float* __restrict__ shift,
                    float* __restrict__ out)
{
    const int gid = blockIdx.x * 256 + threadIdx.x;   // over B*C*N = 2^20
    const int b = gid >> 18;            // C*N = 2^18
    const int o = (gid >> 12) & 63;
    const int n = gid & 4095;
    const int base = o * Ttot + ((b * Npts + n) << 5);   // 128B aligned
    const float4* p = (const float4*)(Y + base);
    const float sc = scale[o];
    const float sh = shift[o];
    float mx = 0.0f;                    // ReLU floor
#pragma unroll
    for (int i = 0; i < 8; ++i) {
        float4 v = p[i];
        mx = fmaxf(mx, v.x * sc + sh);
        mx = fmaxf(mx, v.y * sc + sh);
        mx = fmaxf(mx, v.z * sc + sh);
        mx = fmaxf(mx, v.w * sc + sh);
    }
    out[Bsz * 3 * Npts + gid] = mx;     // feat part after pos1 passthrough
}

// ---------------------------------------------------------------------------
// 9. pos1 passthrough (first tuple element of the reference output).
// ---------------------------------------------------------------------------
__global__ __launch_bounds__(256)
void copy_pos_kernel(const float* __restrict__ pos1, float* __restrict__ out)
{
    const int gid = blockIdx.x * 256 + threadIdx.x;
    out[gid] = pos1[gid];
}

// ---------------------------------------------------------------------------
// Launch
// ---------------------------------------------------------------------------
extern "C" void kernel_launch(void* const* d_in, const int* in_sizes, int n_in,
                              void* d_out, int out_size, void* d_ws, size_t ws_size,
                              hipStream_t stream)
{
    const float* pos1 = (const float*)d_in[0];
    const float* pos2 = (const float*)d_in[1];
    const float* f1   = (const float*)d_in[2];
    const float* f2   = (const float*)d_in[3];
    const float* W0   = (const float*)d_in[4];
    const float* g0   = (const float*)d_in[5];
    const float* b0   = (const float*)d_in[6];
    const float* W1   = (const float*)d_in[7];
    const float* g1   = (const float*)d_in[8];
    const float* b1   = (const float*)d_in[9];
    const float* W2   = (const float*)d_in[10];
    const float* g2   = (const float*)d_in[11];
    const float* b2   = (const float*)d_in[12];
    float* out = (float*)d_out;

    // workspace layout (byte offsets, 256B aligned)
    char* ws = (char*)d_ws;
    const size_t OFF_IDX = 0;                                  // 2 MB
    const size_t OFF_YA  = OFF_IDX + (size_t)Ttot * 4;         // 134 MB
    const size_t OFF_YB  = OFF_YA + (size_t)64 * Ttot * 4;     // 134 MB
    const size_t OFF_SUM = OFF_YB + (size_t)64 * Ttot * 4;     // 4 MB
    const size_t OFF_SQ  = OFF_SUM + (size_t)64 * NBLK * 4;    // 4 MB
    const size_t OFF_BN  = OFF_SQ + (size_t)64 * NBLK * 4;     // 6*64 floats

    int*   idx  = (int*)(ws + OFF_IDX);
    float* YA   = (float*)(ws + OFF_YA);
    float* YB   = (float*)(ws + OFF_YB);
    float* gSum = (float*)(ws + OFF_SUM);
    float* gSq  = (float*)(ws + OFF_SQ);
    float* sc0  = (float*)(ws + OFF_BN);
    float* sh0  = sc0 + 64;
    float* sc1  = sh0 + 64;
    float* sh1  = sc1 + 64;
    float* sc2  = sh1 + 64;
    float* sh2  = sc2 + 64;

    knn_kernel<<<Bsz * Npts / KNN_BLK, KNN_BLK, 0, stream>>>(pos1, pos2, idx);

    mlp_gemm_kernel<0><<<NBLK, 256, 0, stream>>>(
        W0, pos1, pos2, f1, f2, idx, nullptr, nullptr, nullptr, YA, gSum, gSq);
    reduce_stats_kernel<<<64, 256, 0, stream>>>(gSum, gSq, g0, b0, sc0, sh0);

    mlp_gemm_kernel<1><<<NBLK, 256, 0, stream>>>(
        W1, pos1, pos2, f1, f2, idx, YA, sc0, sh0, YB, gSum, gSq);
    reduce_stats_kernel<<<64, 256, 0, stream>>>(gSum, gSq, g1, b1, sc1, sh1);

    mlp_gemm_kernel<2><<<NBLK, 256, 0, stream>>>(
        W2, pos1, pos2, f1, f2, idx, YB, sc1, sh1, YA, gSum, gSq);
    reduce_stats_kernel<<<64, 256, 0, stream>>>(gSum, gSq, g2, b2, sc2, sh2);

    maxpool_kernel<<<(Bsz * Cch * Npts) / 256, 256, 0, stream>>>(YA, sc2, sh2, out);
    copy_pos_kernel<<<(Bsz * 3 * Npts) / 256, 256, 0, stream>>>(pos1, out);
}